// EdgeEncoding_57354993271160
// MI455X (gfx1250) — compile-verified
//
#include <hip/hip_runtime.h>

#define N_NODES  8192
#define N_EDGES  262144
#define N_PATHS  1000000
#define MAX_PATH 5
#define EDGE_DIM 32
#define K_TOTAL  (MAX_PATH * EDGE_DIM)   // 160
#define N_CHUNKS (K_TOTAL / 4)           // 40 K=4 steps
#define N_TILES  (N_PATHS / 16)          // 62500 (exact, no tail)

typedef __attribute__((ext_vector_type(2))) float v2f;
typedef __attribute__((ext_vector_type(4))) float v4f;
typedef __attribute__((ext_vector_type(8))) float v8f;

// ---------------------------------------------------------------------------
// Kernel 1: zero the 8192x8192 f32 output (256 MB). Vectorized b128 stores.
// ---------------------------------------------------------------------------
__global__ void EdgeEnc_zero_kernel(v4f* __restrict__ out, long long n4) {
    long long i      = (long long)blockIdx.x * blockDim.x + threadIdx.x;
    long long stride = (long long)gridDim.x * blockDim.x;
    v4f z = {0.f, 0.f, 0.f, 0.f};
    for (; i < n4; i += stride) out[i] = z;
}

// ---------------------------------------------------------------------------
// Kernel 2: per-path encoding via V_WMMA_F32_16X16X4_F32 + scatter.
//
// One wave handles 16 paths. K = hop*32 + dim (0..159), 40 chunks of K=4.
// A layout (ISA 7.12.2, 32-bit 16x4): lane L holds row M = L%16;
// VGPR0 = K0 (lanes 0-15) / K2 (lanes 16-31), VGPR1 = K1/K3, so each lane
// supplies a contiguous float2 at dim offset (chunk&7)*4 + (lane>=16)*2.
// B holds the weight chunk replicated across all 16 columns -> every D
// column carries the same dot product. Length masking is folded into A via
// a 0/1 multiply (branchless: loads are unconditional and hoistable, EXEC
// stays all-ones for WMMA, and the scheduler can batch loads ahead of the
// accumulator chain instead of load->wait->wmma lockstep).
// ---------------------------------------------------------------------------
__global__ void EdgeEnc_path_kernel(const float* __restrict__ edge_attr,
                                    const float* __restrict__ edge_w,
                                    const int*   __restrict__ paths,
                                    const int*   __restrict__ path_lens,
                                    const int*   __restrict__ src,
                                    const int*   __restrict__ dst,
                                    float*       __restrict__ out) {
    __shared__ float w_lds[K_TOTAL];
    if (threadIdx.x < K_TOTAL) w_lds[threadIdx.x] = edge_w[threadIdx.x];
    __syncthreads();

    const int lane = threadIdx.x & 31;
    const int half = lane >> 4;           // which K-pair this lane supplies
    const int m    = lane & 15;           // path row within the 16-path tile
    const int koff = half << 1;           // +0 or +2 within each K=4 chunk

    // Tile-invariant B fragments: read LDS once per wave (80 VGPRs).
    v2f bv[N_CHUNKS];
#pragma unroll
    for (int chunk = 0; chunk < N_CHUNKS; ++chunk) {
        const int hop   = chunk >> 3;
        const int kbase = ((chunk & 7) << 2) + koff;
        bv[chunk].x = w_lds[hop * EDGE_DIM + kbase];
        bv[chunk].y = w_lds[hop * EDGE_DIM + kbase + 1];
    }

    const int waves_per_grid = (gridDim.x * blockDim.x) >> 5;
    int wave_id = (blockIdx.x * blockDim.x + threadIdx.x) >> 5;

    for (int tile = wave_id; tile < N_TILES; tile += waves_per_grid) {
        const int p = tile * 16 + m;      // < N_PATHS (N_PATHS % 16 == 0)

        int len = path_lens[p];
        if (len < 0) len = 0;
        if (len > MAX_PATH) len = MAX_PATH;

        // Unconditional: path entries are always valid edge indices.
        int eidx[MAX_PATH];
        float fm[MAX_PATH];
#pragma unroll
        for (int l = 0; l < MAX_PATH; ++l) {
            eidx[l] = paths[p * MAX_PATH + l];
            fm[l]   = (l < len) ? 1.0f : 0.0f;
        }

        v8f c = {0.f, 0.f, 0.f, 0.f, 0.f, 0.f, 0.f, 0.f};

#pragma unroll
        for (int hop = 0; hop < MAX_PATH; ++hop) {
            const float* rowp =
                edge_attr + (size_t)eidx[hop] * EDGE_DIM + koff;
            const float msk = fm[hop];
#pragma unroll
            for (int j = 0; j < 8; ++j) {
                v2f a;
                a.x = rowp[j * 4]     * msk;   // merges into global_load_b64
                a.y = rowp[j * 4 + 1] * msk;
                c = __builtin_amdgcn_wmma_f32_16x16x4_f32(
                        /*neg_a=*/false, a, /*neg_b=*/false, bv[hop * 8 + j],
                        /*c_mod=*/(short)0, c,
                        /*reuse_a=*/false, /*reuse_b=*/false);
            }
        }

        // D layout: lanes 0-15 hold rows 0-7 in c[0..7]; lanes 16-31 hold
        // rows 8-15. Lanes {0-7, 16-23} each scatter one path's scalar.
        if ((lane & 8) == 0) {
            const int r = lane & 7;
            float val = c[0];
            if (r == 1) val = c[1];
            else if (r == 2) val = c[2];
            else if (r == 3) val = c[3];
            else if (r == 4) val = c[4];
            else if (r == 5) val = c[5];
            else if (r == 6) val = c[6];
            else if (r == 7) val = c[7];

            const int pr = tile * 16 + half * 8 + r;
            int prl = path_lens[pr];
            if (prl < 0) prl = 0;
            if (prl > MAX_PATH) prl = MAX_PATH;

            float enc = (prl > 0) ? (val / (float)prl) : 0.f;
            if (enc != enc) enc = 0.f;                       // nan_to_num
            else if (enc >  3.4e38f) enc =  3.4e38f;
            else if (enc < -3.4e38f) enc = -3.4e38f;

            const long long o = (long long)src[pr] * N_NODES + dst[pr];
            out[o] = enc;
        }
    }
}

// ---------------------------------------------------------------------------
// Launch. Input order: x, edge_attr, edge_weights, paths, path_lens, src, dst
// ---------------------------------------------------------------------------
extern "C" void kernel_launch(void* const* d_in, const int* in_sizes, int n_in,
                              void* d_out, int out_size, void* d_ws, size_t ws_size,
                              hipStream_t stream) {
    (void)in_sizes; (void)n_in; (void)d_ws; (void)ws_size;

    const float* edge_attr = (const float*)d_in[1];
    const float* edge_w    = (const float*)d_in[2];
    const int*   paths     = (const int*)d_in[3];
    const int*   path_lens = (const int*)d_in[4];
    const int*   src       = (const int*)d_in[5];
    const int*   dst       = (const int*)d_in[6];
    float*       out       = (float*)d_out;

    // 1) zero the 8192x8192 output (256 MB) with b128 stores
    const long long n4 = (long long)out_size / 4;   // 16,777,216 float4
    EdgeEnc_zero_kernel<<<2048, 256, 0, stream>>>((v4f*)out, n4);

    // 2) WMMA path encoding + scatter: 8 waves/block, persistent over tiles
    EdgeEnc_path_kernel<<<1024, 256, 0, stream>>>(edge_attr, edge_w, paths,
                                                  path_lens, src, dst, out);
}